// Sage_74663711473841
// MI455X (gfx1250) — compile-verified
//
#include <hip/hip_runtime.h>

typedef float v2f __attribute__((ext_vector_type(2)));
typedef float v4f __attribute__((ext_vector_type(4)));
typedef float v8f __attribute__((ext_vector_type(8)));

#define N_NODES 100000
#define N_EDGES 1250000
#define D_IN    64
#define D_HID   128
#define D_OUT   64

#define SCAN_N      (N_NODES + 1)
#define SCAN_CHUNK  1024
#define SCAN_BLOCKS ((SCAN_N + SCAN_CHUNK - 1) / SCAN_CHUNK)   // 98

// ---------------------------------------------------------------- CSR build

__global__ __launch_bounds__(256)
void k_zero(int* __restrict__ row_start, int* __restrict__ cursor) {
    int i = blockIdx.x * blockDim.x + threadIdx.x;
    if (i <= N_NODES) row_start[i] = 0;
    if (i <  N_NODES) cursor[i]    = 0;
}

__global__ __launch_bounds__(256)
void k_degree(const int* __restrict__ edge_index, int* __restrict__ row_start) {
    int e = blockIdx.x * blockDim.x + threadIdx.x;
    if (e >= N_EDGES) return;
    int dst = edge_index[N_EDGES + e];            // row 1 of [2,E]
    atomicAdd(&row_start[dst + 1], 1);
}

// inclusive scan of row_start[0..N] -> CSR offsets
__global__ __launch_bounds__(1024)
void k_scan1(const int* __restrict__ data, int* __restrict__ bsum) {
    __shared__ int sh[SCAN_CHUNK];
    int t   = threadIdx.x;
    int idx = blockIdx.x * SCAN_CHUNK + t;
    sh[t] = (idx < SCAN_N) ? data[idx] : 0;
    __syncthreads();
    for (int off = SCAN_CHUNK / 2; off > 0; off >>= 1) {
        int v = 0;
        if (t < off) v = sh[t] + sh[t + off];
        __syncthreads();
        if (t < off) sh[t] = v;
        __syncthreads();
    }
    if (t == 0) bsum[blockIdx.x] = sh[0];
}

__global__ void k_scan2(int* __restrict__ bsum) {   // <<<1,1>>>: exclusive scan of block sums
    int acc = 0;
    for (int i = 0; i < SCAN_BLOCKS; ++i) { int v = bsum[i]; bsum[i] = acc; acc += v; }
}

__global__ __launch_bounds__(1024)
void k_scan3(int* __restrict__ data, const int* __restrict__ bsum) {
    __shared__ int a[SCAN_CHUNK], b[SCAN_CHUNK];
    int t   = threadIdx.x;
    int idx = blockIdx.x * SCAN_CHUNK + t;
    a[t] = (idx < SCAN_N) ? data[idx] : 0;
    __syncthreads();
    int* pin = a; int* pout = b;
    for (int off = 1; off < SCAN_CHUNK; off <<= 1) {
        int s = pin[t];
        if (t >= off) s += pin[t - off];
        pout[t] = s;
        __syncthreads();
        int* tmp = pin; pin = pout; pout = tmp;
    }
    if (idx < SCAN_N) data[idx] = pin[t] + bsum[blockIdx.x];
}

__global__ __launch_bounds__(256)
void k_fill(const int* __restrict__ edge_index, const int* __restrict__ row_start,
            int* __restrict__ cursor, int* __restrict__ csr_src) {
    int e = blockIdx.x * blockDim.x + threadIdx.x;
    if (e >= N_EDGES) return;
    int src = edge_index[e];
    int dst = edge_index[N_EDGES + e];
    int p = atomicAdd(&cursor[dst], 1);
    csr_src[row_start[dst] + p] = src;
}

// ---------------------------------------------------- weight repack (WMMA B layout)
// pk[ks][t][lane] = { W[(4ks+2*half)  *ldw + 16t+l16],
//                     W[(4ks+2*half+1)*ldw + 16t+l16] }   -> one b64/lane per WMMA

__global__ __launch_bounds__(256)
void k_pack(const float* __restrict__ W, float* __restrict__ pk,
            int ldw, int ntiles, int ksteps) {
    int i = blockIdx.x * blockDim.x + threadIdx.x;        // pair index
    int total = ksteps * ntiles * 32;
    if (i >= total) return;
    int lane = i & 31;
    int t    = (i >> 5) % ntiles;
    int ks   = i / (32 * ntiles);
    int half = lane >> 4, l16 = lane & 15;
    int kr   = ks * 4 + 2 * half;
    int col  = t * 16 + l16;
    pk[2 * i]     = W[(size_t)kr * ldw + col];
    pk[2 * i + 1] = W[(size_t)(kr + 1) * ldw + col];
}

// ------------------------------------------------------- mean aggregation
// wave32 per node; 32 neighbor indices loaded at once, broadcast via __shfl
// so row gathers are independent (all in flight). No float atomics.

__global__ __launch_bounds__(256)
void k_agg64(const float* __restrict__ feat, const int* __restrict__ row_start,
             const int* __restrict__ csr_src, float* __restrict__ agg) {
    int node = (blockIdx.x * blockDim.x + threadIdx.x) >> 5;
    int lane = threadIdx.x & 31;
    int s = row_start[node], e = row_start[node + 1];
    float ax = 0.f, ay = 0.f;
    const float* fbase = feat + lane * 2;
    for (int base = s; base < e; base += 32) {
        int idx = 0;
        if (base + lane < e) idx = csr_src[base + lane];
        int n = e - base; if (n > 32) n = 32;
        for (int j = 0; j < n; ++j) {
            int src = __shfl(idx, j, 32);
            v2f v = *(const v2f*)(fbase + (size_t)src * D_IN);
            ax += v.x; ay += v.y;
        }
    }
    float sc = 1.0f / fmaxf((float)(e - s), 1.0f);
    v2f o; o.x = ax * sc; o.y = ay * sc;
    *(v2f*)(agg + (size_t)node * D_IN + lane * 2) = o;
}

__global__ __launch_bounds__(256)
void k_agg128(const float* __restrict__ feat, const int* __restrict__ row_start,
              const int* __restrict__ csr_src, float* __restrict__ agg) {
    int node = (blockIdx.x * blockDim.x + threadIdx.x) >> 5;
    int lane = threadIdx.x & 31;
    int s = row_start[node], e = row_start[node + 1];
    float a0 = 0.f, a1 = 0.f, a2 = 0.f, a3 = 0.f;
    const float* fbase = feat + lane * 4;
    for (int base = s; base < e; base += 32) {
        int idx = 0;
        if (base + lane < e) idx = csr_src[base + lane];
        int n = e - base; if (n > 32) n = 32;
        for (int j = 0; j < n; ++j) {
            int src = __shfl(idx, j, 32);
            v4f v = *(const v4f*)(fbase + (size_t)src * D_HID);
            a0 += v.x; a1 += v.y; a2 += v.z; a3 += v.w;
        }
    }
    float sc = 1.0f / fmaxf((float)(e - s), 1.0f);
    v4f o; o.x = a0 * sc; o.y = a1 * sc; o.z = a2 * sc; o.w = a3 * sc;
    *(v4f*)(agg + (size_t)node * D_HID + lane * 4) = o;
}

// ------------------------------------------------------------ WMMA GEMMs
// h = relu(agg1 @ W1_l + b1 + x @ W1_r); A:16x4 f32 (b64/lane), B packed (b64/lane)

__global__ __launch_bounds__(64)
void k_gemm1(const float* __restrict__ agg, const float* __restrict__ x,
             const float* __restrict__ pkL, const float* __restrict__ pkR,
             const float* __restrict__ b1, float* __restrict__ h) {
    const int wave = (blockIdx.x * blockDim.x + threadIdx.x) >> 5;   // 0..6249
    const int lane = threadIdx.x & 31;
    const int half = lane >> 4;
    const int l16  = lane & 15;
    const int row0 = wave * 16;

    v8f acc[8];
#pragma unroll
    for (int t = 0; t < 8; ++t)
#pragma unroll
        for (int j = 0; j < 8; ++j) acc[t][j] = 0.0f;

    const float* aRow = agg + (size_t)(row0 + l16) * D_IN + 2 * half;
    const float* xRow = x   + (size_t)(row0 + l16) * D_IN + 2 * half;
    const v2f*   bL   = (const v2f*)pkL + lane;
    const v2f*   bR   = (const v2f*)pkR + lane;

    for (int kk = 0; kk < D_IN; kk += 4) {
        v2f aA = *(const v2f*)(aRow + kk);
        v2f aX = *(const v2f*)(xRow + kk);
        const int ksOff = (kk >> 2) * 8 * 32;
#pragma unroll
        for (int t = 0; t < 8; ++t) {
            v2f wl = bL[ksOff + t * 32];
            acc[t] = __builtin_amdgcn_wmma_f32_16x16x4_f32(
                false, aA, false, wl, (short)0, acc[t], false, false);
            v2f wr = bR[ksOff + t * 32];
            acc[t] = __builtin_amdgcn_wmma_f32_16x16x4_f32(
                false, aX, false, wr, (short)0, acc[t], false, false);
        }
    }

#pragma unroll
    for (int t = 0; t < 8; ++t) {
        const int col  = t * 16 + l16;
        const float bb = b1[col];
#pragma unroll
        for (int j = 0; j < 8; ++j) {
            const int r = row0 + j + 8 * half;                 // C layout: VGPR j -> M=j / j+8
            h[(size_t)r * D_HID + col] = fmaxf(acc[t][j] + bb, 0.0f);
        }
    }
}

// out = agg2 @ W2_l + b2 + h @ W2_r
__global__ __launch_bounds__(64)
void k_gemm2(const float* __restrict__ agg, const float* __restrict__ hIn,
             const float* __restrict__ pkL, const float* __restrict__ pkR,
             const float* __restrict__ b2, float* __restrict__ out) {
    const int wave = (blockIdx.x * blockDim.x + threadIdx.x) >> 5;
    const int lane = threadIdx.x & 31;
    const int half = lane >> 4;
    const int l16  = lane & 15;
    const int row0 = wave * 16;

    v8f acc[4];
#pragma unroll
    for (int t = 0; t < 4; ++t)
#pragma unroll
        for (int j = 0; j < 8; ++j) acc[t][j] = 0.0f;

    const float* aRow = agg + (size_t)(row0 + l16) * D_HID + 2 * half;
    const float* hRow = hIn + (size_t)(row0 + l16) * D_HID + 2 * half;
    const v2f*   bL   = (const v2f*)pkL + lane;
    const v2f*   bR   = (const v2f*)pkR + lane;

    for (int kk = 0; kk < D_HID; kk += 4) {
        v2f aA = *(const v2f*)(aRow + kk);
        v2f aH = *(const v2f*)(hRow + kk);
        const int ksOff = (kk >> 2) * 4 * 32;
#pragma unroll
        for (int t = 0; t < 4; ++t) {
            v2f wl = bL[ksOff + t * 32];
            acc[t] = __builtin_amdgcn_wmma_f32_16x16x4_f32(
                false, aA, false, wl, (short)0, acc[t], false, false);
            v2f wr = bR[ksOff + t * 32];
            acc[t] = __builtin_amdgcn_wmma_f32_16x16x4_f32(
                false, aH, false, wr, (short)0, acc[t], false, false);
        }
    }

#pragma unroll
    for (int t = 0; t < 4; ++t) {
        const int col  = t * 16 + l16;
        const float bb = b2[col];
#pragma unroll
        for (int j = 0; j < 8; ++j) {
            const int r = row0 + j + 8 * half;
            out[(size_t)r * D_OUT + col] = acc[t][j] + bb;
        }
    }
}

// ---------------------------------------------------------------- launcher

static inline size_t alignUp(size_t v, size_t a) { return (v + a - 1) & ~(a - 1); }

extern "C" void kernel_launch(void* const* d_in, const int* in_sizes, int n_in,
                              void* d_out, int out_size, void* d_ws, size_t ws_size,
                              hipStream_t stream) {
    (void)in_sizes; (void)n_in; (void)out_size; (void)ws_size;

    const float* x    = (const float*)d_in[0];   // [N, 64]
    const int*   ei   = (const int*)  d_in[1];   // [2, E]
    const float* W1l  = (const float*)d_in[2];   // [64, 128]
    const float* b1   = (const float*)d_in[3];   // [128]
    const float* W1r  = (const float*)d_in[4];   // [64, 128]
    const float* W2l  = (const float*)d_in[5];   // [128, 64]
    const float* b2   = (const float*)d_in[6];   // [64]
    const float* W2r  = (const float*)d_in[7];   // [128, 64]
    float*       outp = (float*)d_out;           // [N, 64]

    // workspace carve-up (~134 MB)
    char* ws = (char*)d_ws;
    size_t off = 0;
    int* row_start = (int*)(ws + off); off = alignUp(off + sizeof(int) * (N_NODES + 1), 256);
    int* cursor    = (int*)(ws + off); off = alignUp(off + sizeof(int) * N_NODES, 256);
    int* bsum      = (int*)(ws + off); off = alignUp(off + sizeof(int) * SCAN_BLOCKS, 256);
    int* csr_src   = (int*)(ws + off); off = alignUp(off + sizeof(int) * N_EDGES, 256);
    float* pk1l    = (float*)(ws + off); off = alignUp(off + sizeof(float) * 16 * 8 * 64, 256);
    float* pk1r    = (float*)(ws + off); off = alignUp(off + sizeof(float) * 16 * 8 * 64, 256);
    float* pk2l    = (float*)(ws + off); off = alignUp(off + sizeof(float) * 32 * 4 * 64, 256);
    float* pk2r    = (float*)(ws + off); off = alignUp(off + sizeof(float) * 32 * 4 * 64, 256);
    float* agg1    = (float*)(ws + off); off = alignUp(off + sizeof(float) * (size_t)N_NODES * D_IN, 256);
    float* agg2    = (float*)(ws + off); off = alignUp(off + sizeof(float) * (size_t)N_NODES * D_HID, 256);
    float* hbuf    = (float*)(ws + off); off = alignUp(off + sizeof(float) * (size_t)N_NODES * D_HID, 256);

    const int edgeBlocks = (N_EDGES + 255) / 256;
    const int zeroBlocks = (N_NODES + 1 + 255) / 256;
    const int aggBlocks  = (N_NODES * 32) / 256;         // 12500, exact
    const int gemmBlocks = (N_NODES / 16) / 2;           // 3125 blocks x 2 waves, exact
    const int packPairs  = 16 * 8 * 32;                  // 4096 per weight matrix
    const int packBlocks = (packPairs + 255) / 256;

    // weight repack into WMMA-native B layout (tiny, once per launch)
    k_pack<<<packBlocks, 256, 0, stream>>>(W1l, pk1l, D_HID, 8, 16);
    k_pack<<<packBlocks, 256, 0, stream>>>(W1r, pk1r, D_HID, 8, 16);
    k_pack<<<packBlocks, 256, 0, stream>>>(W2l, pk2l, D_OUT, 4, 32);
    k_pack<<<packBlocks, 256, 0, stream>>>(W2r, pk2r, D_OUT, 4, 32);

    // CSR build (int atomics only; ~2.5M atomics total)
    k_zero  <<<zeroBlocks, 256, 0, stream>>>(row_start, cursor);
    k_degree<<<edgeBlocks, 256, 0, stream>>>(ei, row_start);
    k_scan1 <<<SCAN_BLOCKS, SCAN_CHUNK, 0, stream>>>(row_start, bsum);
    k_scan2 <<<1, 1, 0, stream>>>(bsum);
    k_scan3 <<<SCAN_BLOCKS, SCAN_CHUNK, 0, stream>>>(row_start, bsum);
    k_fill  <<<edgeBlocks, 256, 0, stream>>>(ei, row_start, cursor, csr_src);

    // layer 1
    k_agg64 <<<aggBlocks, 256, 0, stream>>>(x, row_start, csr_src, agg1);
    k_gemm1 <<<gemmBlocks, 64, 0, stream>>>(agg1, x, pk1l, pk1r, b1, hbuf);

    // layer 2
    k_agg128<<<aggBlocks, 256, 0, stream>>>(hbuf, row_start, csr_src, agg2);
    k_gemm2 <<<gemmBlocks, 64, 0, stream>>>(agg2, hbuf, pk2l, pk2r, b2, outp);
}